// ResNetFeatureExtractor_proposedStructuredPooling_1709396983953
// MI455X (gfx1250) — compile-verified
//
#include <hip/hip_runtime.h>
#include <hip/hip_bf16.h>

// ---------------------------------------------------------------------------
// S[b,r,c] = (1/49) * sum_k Mp[b,r,k] * Fr[b,c,k]   (rank-7 factorization of
// the brik,bckj->brc einsum).  Memory-bound: ~88MB traffic -> ~3.8us @23.3TB/s
// on MI455X HBM.  fp32 WMMA (V_WMMA_F32_16X16X4_F32) does the contractions
// exactly.  Operands are stored K=8-padded (zeros) so the WMMA kernels are
// branch-free: unconditional global_load_b64 fragment loads, uniform store
// guards only on the single partial row-tile.
// ---------------------------------------------------------------------------

typedef __attribute__((ext_vector_type(2))) float v2f;
typedef __attribute__((ext_vector_type(8))) float v8f;

#define B_ 64
#define R_ 100
#define RP 112          // R padded to multiple of 16 (7 row-tiles)
#define C_ 2048
#define P_ 7
#define K8 8            // K padded to 8 (two K=4 WMMA steps)
#define PP 49
#define H_ 512

// ---- K1: Fr8[b,c,k] = sum_j feats[b,c,k,j] (k<7), 0 for k==7 --------------
__global__ void k_reduce_feats(const float* __restrict__ feats,
                               float* __restrict__ Fr8) {
  int idx = blockIdx.x * blockDim.x + threadIdx.x;   // over B*C*8
  const int total = B_ * C_ * K8;
  if (idx >= total) return;
  int k  = idx & 7;
  int bc = idx >> 3;
  float s = 0.f;
  if (k < P_) {
    const float* p = feats + (size_t)bc * PP + (size_t)k * P_;
#pragma unroll
    for (int j = 0; j < P_; ++j) s += p[j];
  }
  Fr8[idx] = s;
}

// ---- K2: Mp8[b,r,k] = sum_i masks[b,r,i,k]; zero pad rows & k==7 ----------
__global__ void k_reduce_masks(const float* __restrict__ masks,
                               float* __restrict__ Mp8) {
  int idx = blockIdx.x * blockDim.x + threadIdx.x;   // over B*RP*8
  const int total = B_ * RP * K8;
  if (idx >= total) return;
  int k  = idx & 7;
  int br = idx >> 3;                 // b*RP + r
  int b  = br / RP;
  int r  = br % RP;
  float s = 0.f;
  if (r < R_ && k < P_) {
    const float* p = masks + ((size_t)b * R_ + r) * PP + k;
#pragma unroll
    for (int i = 0; i < P_; ++i) s += p[i * P_];
  }
  Mp8[idx] = s;
}

// ---- K2b: Mbar[b,k] = sum_r Mp8[b,r,k]  (pad rows are 0) ------------------
__global__ void k_mbar(const float* __restrict__ Mp8, float* __restrict__ Mbar) {
  int idx = blockIdx.x * blockDim.x + threadIdx.x;   // over B*8
  if (idx >= B_ * K8) return;
  int b = idx >> 3, k = idx & 7;
  float s = 0.f;
  for (int r = 0; r < RP; ++r) s += Mp8[((size_t)b * RP + r) * K8 + k];
  Mbar[idx] = s;
}

// ---- K3: Sm[b,c] = (1/(49*R)) sum_k Mbar[b,k]*Fr8[b,c,k] ------------------
__global__ void k_sm(const float* __restrict__ Fr8,
                     const float* __restrict__ Mbar,
                     float* __restrict__ Sm) {
  int idx = blockIdx.x * blockDim.x + threadIdx.x;   // over B*C
  if (idx >= B_ * C_) return;
  int b = idx / C_;
  const float* f  = Fr8 + (size_t)idx * K8;
  const float* mb = Mbar + b * K8;
  float s = 0.f;
#pragma unroll
  for (int k = 0; k < K8; ++k) s += mb[k] * f[k];    // k==7 term is 0*0
  Sm[idx] = s * (1.0f / (PP * R_));
}

// ---- K4: S[b] = (1/49) Mp8[b](RP x 8) * Fr8[b]^T(8 x C), fp32 WMMA K=4 ----
// One wave per 16x16 output tile; branch-free fragment loads (K padded),
// wave-uniform store guard (only mt==6 is a partial row-tile).
__global__ void k_s_wmma(const float* __restrict__ Mp8,
                         const float* __restrict__ Fr8,
                         float* __restrict__ S) {
  const int MT = RP / 16;                   // 7
  const int NT = C_ / 16;                   // 128
  int wave = blockIdx.x * (blockDim.x >> 5) + (threadIdx.x >> 5);
  int b  = wave / (MT * NT);
  int t  = wave % (MT * NT);
  int mt = t / NT, nt = t % NT;

  int lane = threadIdx.x & 31;
  int half = lane >> 4;                     // upper half-wave: K += 2
  int lm   = lane & 15;

  // A frag: row = mt*16+lm (padded rows are zero), K pair at 2*half
  const float* ap = Mp8 + ((size_t)b * RP + mt * 16 + lm) * K8 + 2 * half;
  // B frag: col = nt*16+lm
  const float* bp = Fr8 + ((size_t)b * C_ + nt * 16 + lm) * K8 + 2 * half;

  v2f a0 = *(const v2f*)(ap);
  v2f b0 = *(const v2f*)(bp);
  v2f a1 = *(const v2f*)(ap + 4);
  v2f b1 = *(const v2f*)(bp + 4);

  v8f acc = {};
  acc = __builtin_amdgcn_wmma_f32_16x16x4_f32(false, a0, false, b0,
                                              (short)0, acc, false, false);
  acc = __builtin_amdgcn_wmma_f32_16x16x4_f32(false, a1, false, b1,
                                              (short)0, acc, false, false);

  const float inv = 1.0f / PP;
  float* out = S + ((size_t)b * R_ + mt * 16 + half * 8) * C_ + nt * 16 + lm;
  if (mt < MT - 1) {                        // full tile: unguarded NT stores
#pragma unroll
    for (int v = 0; v < 8; ++v)
      __builtin_nontemporal_store(acc[v] * inv, out + (size_t)v * C_);
  } else {                                  // rows 96..111: only 96..99 valid
#pragma unroll
    for (int v = 0; v < 8; ++v) {
      int m = 96 + half * 8 + v;
      if (m < R_)
        __builtin_nontemporal_store(acc[v] * inv, out + (size_t)v * C_);
    }
  }
}

// ---- K5: h0 = Sm@W1^T + b1 ; c0 = Sm@W2^T + b2, fp32 WMMA K-loop ----------
// 256 waves: wave 0..127 -> h0 tiles, 128..255 -> c0 tiles (M=64,N=512,K=2048)
__global__ void k_head_wmma(const float* __restrict__ Sm,
                            const float* __restrict__ W1,
                            const float* __restrict__ b1,
                            const float* __restrict__ W2,
                            const float* __restrict__ b2,
                            float* __restrict__ h0,
                            float* __restrict__ c0) {
  const int MT = B_ / 16;                   // 4
  const int NT = H_ / 16;                   // 32
  int wave  = blockIdx.x * (blockDim.x >> 5) + (threadIdx.x >> 5);
  int which = wave / (MT * NT);             // 0 = h0, 1 = c0 (wave-uniform)
  int t     = wave % (MT * NT);
  int mt = t / NT, nt = t % NT;

  const float* W    = which ? W2 : W1;
  const float* bias = which ? b2 : b1;
  float*       out  = which ? c0 : h0;

  int lane = threadIdx.x & 31;
  int half = lane >> 4;
  int lm   = lane & 15;

  const float* ap = Sm + (size_t)(mt * 16 + lm) * C_ + 2 * half;
  const float* bp = W  + (size_t)(nt * 16 + lm) * C_ + 2 * half;

  v8f acc = {};
#pragma unroll 8
  for (int kb = 0; kb < C_; kb += 4) {      // 512 K-steps, 8 WMMAs per body
    v2f a  = *(const v2f*)(ap + kb);
    v2f bf = *(const v2f*)(bp + kb);
    acc = __builtin_amdgcn_wmma_f32_16x16x4_f32(false, a, false, bf,
                                               (short)0, acc, false, false);
  }

  float bv = bias[nt * 16 + lm];
#pragma unroll
  for (int v = 0; v < 8; ++v) {
    int m = mt * 16 + half * 8 + v;         // batch row
    out[(size_t)m * H_ + nt * 16 + lm] = acc[v] + bv;
  }
}

// ---------------------------------------------------------------------------
extern "C" void kernel_launch(void* const* d_in, const int* in_sizes, int n_in,
                              void* d_out, int out_size, void* d_ws, size_t ws_size,
                              hipStream_t stream) {
  const float* feats = (const float*)d_in[0];   // (B,C,P,P)
  const float* masks = (const float*)d_in[1];   // (B,R,P,P)
  const float* W1    = (const float*)d_in[2];   // (H,C)
  const float* b1    = (const float*)d_in[3];   // (H,)
  const float* W2    = (const float*)d_in[4];   // (H,C)
  const float* b2    = (const float*)d_in[5];   // (H,)

  float* out = (float*)d_out;
  float* S   = out;                                  // B*R*C
  float* h0  = out + (size_t)B_ * R_ * C_;           // B*H
  float* c0  = h0 + (size_t)B_ * H_;                 // B*H

  // workspace layout (floats): Fr8 | Mp8 | Mbar | Sm   (~5 MB total)
  float* ws   = (float*)d_ws;
  float* Fr8  = ws;                                  // B*C*8  = 1048576
  float* Mp8  = Fr8 + (size_t)B_ * C_ * K8;          // B*RP*8 = 57344
  float* Mbar = Mp8 + (size_t)B_ * RP * K8;          // B*8    = 512
  float* Sm   = Mbar + (size_t)B_ * K8;              // B*C    = 131072

  // K1: feats row-sum -> K=8 padded
  {
    int total = B_ * C_ * K8;
    k_reduce_feats<<<(total + 255) / 256, 256, 0, stream>>>(feats, Fr8);
  }
  // K2: masks col-sum -> row+K padded
  {
    int total = B_ * RP * K8;
    k_reduce_masks<<<(total + 255) / 256, 256, 0, stream>>>(masks, Mp8);
  }
  // K2b: Mbar
  k_mbar<<<2, 256, 0, stream>>>(Mp8, Mbar);
  // K3: Sm
  {
    int total = B_ * C_;
    k_sm<<<(total + 255) / 256, 256, 0, stream>>>(Fr8, Mbar, Sm);
  }
  // K4: batched WMMA GEMM -> S.  waves = 64 * 7 * 128 = 57344, 8 waves/block
  {
    int waves = B_ * (RP / 16) * (C_ / 16);
    k_s_wmma<<<waves / 8, 256, 0, stream>>>(Mp8, Fr8, S);
  }
  // K5: head GEMMs -> h0, c0.  256 waves, 8 waves/block
  k_head_wmma<<<32, 256, 0, stream>>>(Sm, W1, b1, W2, b2, h0, c0);
}